// ManifoldEmbed_80994493268366
// MI455X (gfx1250) — compile-verified
//
#include <hip/hip_runtime.h>
#include <hip/hip_bf16.h>

typedef __attribute__((ext_vector_type(2))) float v2f;
typedef __attribute__((ext_vector_type(8))) float v8f;

#define NROWS 32768
#define DDIM  4096
#define MDIM  4
#define NCH   16
#define KDIM  64   // C*M

// ---------------------------------------------------------------------------
// Kernel 1: coords = x @ encoder_w.T  (wave per row), softmax gate over 16
// chart centers, emit a[n, c*4+m] = gate[n,c]*coords[n,m]  -> ws (N x 64 f32)
// ---------------------------------------------------------------------------
__global__ __launch_bounds__(256) void k_encode(const float* __restrict__ x,
                                                const float* __restrict__ ew,
                                                const float* __restrict__ cc,
                                                float* __restrict__ a) {
    __shared__ float lds_e[MDIM * DDIM];          // 64 KB: encoder rows
    for (int i = threadIdx.x; i < (MDIM * DDIM) / 4; i += 256) {
        ((float4*)lds_e)[i] = ((const float4*)ew)[i];
    }
    __syncthreads();

    const int lane = threadIdx.x & 31;
    const int wave = threadIdx.x >> 5;
    const int n    = blockIdx.x * 8 + wave;       // one wave per row
    const float* xrow = x + (size_t)n * DDIM;

    float c0 = 0.f, c1 = 0.f, c2 = 0.f, c3 = 0.f;
    #pragma unroll 4
    for (int i = 0; i < DDIM / (32 * 4); ++i) {   // 32 iters
        const int d = (i * 32 + lane) * 4;
        const float4 xv = *(const float4*)(xrow + d);
        const float4 e0 = *(const float4*)(lds_e + 0 * DDIM + d);
        const float4 e1 = *(const float4*)(lds_e + 1 * DDIM + d);
        const float4 e2 = *(const float4*)(lds_e + 2 * DDIM + d);
        const float4 e3 = *(const float4*)(lds_e + 3 * DDIM + d);
        c0 += xv.x * e0.x + xv.y * e0.y + xv.z * e0.z + xv.w * e0.w;
        c1 += xv.x * e1.x + xv.y * e1.y + xv.z * e1.z + xv.w * e1.w;
        c2 += xv.x * e2.x + xv.y * e2.y + xv.z * e2.z + xv.w * e2.w;
        c3 += xv.x * e3.x + xv.y * e3.y + xv.z * e3.z + xv.w * e3.w;
    }
    #pragma unroll
    for (int m = 16; m >= 1; m >>= 1) {
        c0 += __shfl_xor(c0, m, 32);
        c1 += __shfl_xor(c1, m, 32);
        c2 += __shfl_xor(c2, m, 32);
        c3 += __shfl_xor(c3, m, 32);
    }

    const int c = lane & 15;
    const float4 ctr = ((const float4*)cc)[c];
    const float d0 = c0 - ctr.x, d1 = c1 - ctr.y, d2 = c2 - ctr.z, d3 = c3 - ctr.w;
    const float dist = sqrtf(d0 * d0 + d1 * d1 + d2 * d2 + d3 * d3);

    float t = -dist;
    float tmax = t;
    #pragma unroll
    for (int m = 8; m >= 1; m >>= 1) tmax = fmaxf(tmax, __shfl_xor(tmax, m, 32));
    float g = __expf(t - tmax);
    float gsum = g;
    #pragma unroll
    for (int m = 8; m >= 1; m >>= 1) gsum += __shfl_xor(gsum, m, 32);
    g /= gsum;

    const float gc = __shfl(g, lane >> 1, 32);
    float2 av;
    av.x = gc * ((lane & 1) ? c2 : c0);
    av.y = gc * ((lane & 1) ? c3 : c1);
    *(float2*)(a + (size_t)n * KDIM + 2 * lane) = av;
}

// ---------------------------------------------------------------------------
// Kernel 2: pack decoder_w into fragment-native layout
//   w2p[colTile][kk][hi][colLocal][pair] , k = kk*4 + hi*2 + pair
//   value = decoder_w[k>>2, colTile*64+colLocal, k&3]
// 64 x 4096 f32 = 1 MB total; one 16 KB slab per column tile.
// ---------------------------------------------------------------------------
__global__ __launch_bounds__(256) void k_pack(const float* __restrict__ dw,
                                              float* __restrict__ w2p) {
    const int t = blockIdx.x * 256 + threadIdx.x;     // 0 .. 64*4096-1
    const int p        =  t        & 1;
    const int colLocal = (t >> 1)  & 63;
    const int hi       = (t >> 7)  & 1;
    const int kk       = (t >> 8)  & 15;
    const int colTile  =  t >> 12;
    const int k = kk * 4 + hi * 2 + p;
    const int d = colTile * 64 + colLocal;
    w2p[t] = dw[(size_t)(k >> 2) * (DDIM * MDIM) + (size_t)d * MDIM + (k & 3)];
}

// ---------------------------------------------------------------------------
// Kernel 3: out[N, D] = a[N,64] @ W2[64, D]  via V_WMMA_F32_16X16X4_F32.
// Block = 8 waves sharing one 64-col slab; B slab (16 KB) staged in LDS once,
// fragments served as single ds_load_b64. Each wave: 16 rows x 64 cols,
// 64 WMMAs (4 tiles x 16 K-steps).
//   A 16x4 f32:  VGPR0 = {K=0 | K=2}, VGPR1 = {K=1 | K=3}, rows on lanes%16
//   B 4x16 f32:  VGPR0 = {K=0 | K=2}, VGPR1 = {K=1 | K=3}, cols on lanes%16
//   C/D 16x16:   VGPR v = row (v | v+8) across half-waves, cols on lanes%16
// ---------------------------------------------------------------------------
__global__ __launch_bounds__(256) void k_gemm(const float* __restrict__ a,
                                              const float* __restrict__ w2p,
                                              float* __restrict__ out) {
    __shared__ float ldsB[16 * 2 * 64 * 2];          // 16 KB B slab

    const int lane     = threadIdx.x & 31;
    const int wave     = threadIdx.x >> 5;
    const int colTile  = blockIdx.x >> 8;            // 64 column slabs
    const int rowTile  = ((blockIdx.x & 255) << 3) + wave;   // 2048 row tiles
    const int rowBase  = rowTile * 16;
    const int colBase  = colTile * 64;
    const int r  = lane & 15;
    const int hi = lane >> 4;

    // stage this column slab's packed B into LDS (contiguous, coalesced)
    const float4* bsrc = (const float4*)(w2p + (size_t)colTile * (KDIM * 64));
    for (int i = threadIdx.x; i < (KDIM * 64) / 4; i += 256) {
        ((float4*)ldsB)[i] = bsrc[i];
    }
    __syncthreads();

    // A fragments for all 16 K-steps (float2 per lane each)
    v2f afrag[16];
    const float* arow = a + (size_t)(rowBase + r) * KDIM + hi * 2;
    #pragma unroll
    for (int kk = 0; kk < 16; ++kk)
        afrag[kk] = *(const v2f*)(arow + kk * 4);

    #pragma unroll
    for (int s = 0; s < 4; ++s) {
        v8f acc = {0.f, 0.f, 0.f, 0.f, 0.f, 0.f, 0.f, 0.f};
        const float* bfr = ldsB + ((size_t)hi * 64 + s * 16 + r) * 2;
        #pragma unroll
        for (int kk = 0; kk < 16; ++kk) {
            const v2f bfrag = *(const v2f*)(bfr + (size_t)kk * 256);  // [kk][hi][col][pair]
            acc = __builtin_amdgcn_wmma_f32_16x16x4_f32(
                /*neg_a=*/false, afrag[kk], /*neg_b=*/false, bfrag,
                /*c_mod=*/(short)0, acc, /*reuse_a=*/false, /*reuse_b=*/false);
        }
        float* o = out + (size_t)(rowBase + hi * 8) * DDIM + colBase + s * 16 + r;
        #pragma unroll
        for (int v = 0; v < 8; ++v)
            o[(size_t)v * DDIM] = acc[v];
    }
}

// ---------------------------------------------------------------------------
extern "C" void kernel_launch(void* const* d_in, const int* in_sizes, int n_in,
                              void* d_out, int out_size, void* d_ws, size_t ws_size,
                              hipStream_t stream) {
    const float* x  = (const float*)d_in[0];   // [N, D]
    const float* ew = (const float*)d_in[1];   // [M, D]
    const float* dw = (const float*)d_in[2];   // [C, D, M]
    const float* cc = (const float*)d_in[3];   // [C, M]
    float* out = (float*)d_out;                // [N, D]

    float* a   = (float*)d_ws;                                     // N*64 f32 = 8 MB
    float* w2p = (float*)((char*)d_ws + (size_t)NROWS * KDIM * 4); // 64*D f32 = 1 MB

    k_pack  <<<(KDIM * DDIM) / 256, 256, 0, stream>>>(dw, w2p);
    k_encode<<<NROWS / 8,           256, 0, stream>>>(x, ew, cc, a);
    k_gemm  <<<(NROWS / 16) * (DDIM / 64) / 8, 256, 0, stream>>>(a, w2p, out);
}